// Summarizer_20512763806366
// MI455X (gfx1250) — compile-verified
//
#include <hip/hip_runtime.h>
#include <cstdint>

// Problem constants from the reference
#define N_S   32768           // samples per event
#define HOP_  256
#define NE    4               // events per batch
#define TPB   512             // 16 wave32s
#define CHF   2048            // floats per chunk (TPB * 4, float4 per thread)
#define NCH   (N_S / CHF)     // 16 chunks

// out[b, j] = sum_e (j >= off_e ? x[b, e, j - off_e] : 0),  off_e = idx[b,e]*HOP
// then out[b, :] /= (max_j out[b, j] + 1e-8)
__global__ __launch_bounds__(TPB)
void scatter_add_norm_kernel(const float* __restrict__ x,
                             const int* __restrict__ indices,
                             float* __restrict__ out)
{
    const int b   = blockIdx.x;
    const int tid = threadIdx.x;

    // Double-buffered async staging: 2 stages x 4 events x 2048 floats = 64 KB
    __shared__ float smem[2][NE][CHF];

    // Per-block-uniform shift offsets (indices are uniform within a block)
    int off[NE];
#pragma unroll
    for (int e = 0; e < NE; ++e) off[e] = indices[b * NE + e] * HOP_;

    const float* xb = x + (size_t)b * NE * N_S;
    float*       ob = out + (size_t)b * N_S;

    // Low 32 bits of a generic pointer into LDS == LDS byte offset
    const unsigned lds_base = (unsigned)(uintptr_t)&smem[0][0][0];

    // Issue 4 async b128 loads (one per event) for chunk c into stage st.
    // Each lane targets its own LDS slot -> no cross-lane deps, no barrier.
    auto issue_chunk = [&](int c, int st) {
        const int j4 = c * CHF + tid * 4;          // first of this thread's 4 samples
#pragma unroll
        for (int e = 0; e < NE; ++e) {
            int src = j4 - off[e];
            src = src < 0 ? 0 : src;               // clamp; masked at consume
            const unsigned voff  = (unsigned)src * 4u;                      // byte offset
            const unsigned laddr = lds_base + ((st * NE + e) * CHF + tid * 4) * 4u;
            const float*   base  = xb + (size_t)e * N_S;                    // uniform -> SGPR pair
            asm volatile("global_load_async_to_lds_b128 %0, %1, %2"
                         :: "v"(laddr), "v"(voff), "s"(base)
                         : "memory");
        }
    };

    float4 vals[NCH];
    float  mx = -__builtin_inff();

    issue_chunk(0, 0);

#pragma unroll
    for (int c = 0; c < NCH; ++c) {
        const int st = c & 1;
        if (c + 1 < NCH) {
            // Drain outstanding LDS reads before async engine overwrites stage st^1's peer
            asm volatile("s_wait_dscnt 0x0" ::: "memory");
            issue_chunk(c + 1, st ^ 1);
            asm volatile("s_wait_asynccnt 0x4" ::: "memory");  // chunk c's 4 loads done
        } else {
            asm volatile("s_wait_asynccnt 0x0" ::: "memory");
        }

        const int j4 = c * CHF + tid * 4;
        float4 s = make_float4(0.f, 0.f, 0.f, 0.f);
#pragma unroll
        for (int e = 0; e < NE; ++e) {
            const float4 v = *reinterpret_cast<const float4*>(&smem[st][e][tid * 4]);
            // off[e] is a multiple of 256 -> whole float4 is in- or out-of-range
            if (j4 >= off[e]) { s.x += v.x; s.y += v.y; s.z += v.z; s.w += v.w; }
        }
        vals[c] = s;
        mx = fmaxf(mx, fmaxf(fmaxf(s.x, s.y), fmaxf(s.z, s.w)));
    }

    // ---- block max reduction (wave32 shuffle + LDS across 16 waves) ----
#pragma unroll
    for (int o = 16; o > 0; o >>= 1) mx = fmaxf(mx, __shfl_xor(mx, o, 32));

    float* red = &smem[0][0][0];   // staging no longer needed; asynccnt == 0
    const int wave = tid >> 5;
    const int lane = tid & 31;
    __syncthreads();               // everyone done reading staging
    if (lane == 0) red[wave] = mx;
    __syncthreads();
    if (tid < 32) {
        float m = (tid < (TPB / 32)) ? red[tid] : -__builtin_inff();
#pragma unroll
        for (int o = 16; o > 0; o >>= 1) m = fmaxf(m, __shfl_xor(m, o, 32));
        if (tid == 0) red[0] = m;
    }
    __syncthreads();
    const float scale = 1.0f / (red[0] + 1e-8f);

    // ---- scale from registers and store as b128 ----
#pragma unroll
    for (int c = 0; c < NCH; ++c) {
        const int j4 = c * CHF + tid * 4;
        float4 s = vals[c];
        s.x *= scale; s.y *= scale; s.z *= scale; s.w *= scale;
        *reinterpret_cast<float4*>(&ob[j4]) = s;
    }
}

extern "C" void kernel_launch(void* const* d_in, const int* in_sizes, int n_in,
                              void* d_out, int out_size, void* d_ws, size_t ws_size,
                              hipStream_t stream) {
    const float* x       = (const float*)d_in[0];   // (256, 4, 32768) fp32
    const int*   indices = (const int*)d_in[1];     // (256, 4) int32
    float*       out     = (float*)d_out;           // (256, 1, 32768) fp32

    const int batch = in_sizes[1] / NE;             // 256
    scatter_add_norm_kernel<<<batch, TPB, 0, stream>>>(x, indices, out);
}